// LaplacianPyramidLoss_10222022164596
// MI455X (gfx1250) — compile-verified
//
#include <hip/hip_runtime.h>
#include <cstdint>

// ---------------------------------------------------------------------------
// Laplacian pyramid L1 loss, MI455X (gfx1250, wave32).
// Exploits linearity: pyr(x)-pyr(t) == pyr(x-t). Memory-bound (~580MB total
// traffic -> ~25us floor at 23.3 TB/s). Separable 5-tap Gaussian + stride-2
// downsample expressed as banded matmuls on V_WMMA_F32_16X16X4_F32; tile
// streaming uses gfx1250 async global->LDS loads (ASYNCcnt).
// ---------------------------------------------------------------------------

typedef float v2f __attribute__((ext_vector_type(2)));
typedef float v8f __attribute__((ext_vector_type(8)));

#define DEVINL __device__ __forceinline__

// ---- gfx1250 async global->LDS copy (4B per lane) --------------------------
DEVINL void async_cp_b32(const float* gsrc, const float* lds_dst) {
#if __has_builtin(__builtin_amdgcn_global_load_async_to_lds_b32)
  // Builtin takes AS-qualified int*: AS1 keeps the 64-bit flat address; AS3 is
  // the low 32 bits of the flat LDS address (aperture lives in the high bits).
  __builtin_amdgcn_global_load_async_to_lds_b32(
      (__attribute__((address_space(1))) int*)(uintptr_t)gsrc,
      (__attribute__((address_space(3))) int*)(uint32_t)(uintptr_t)lds_dst,
      0, 0);
#else
  asm volatile("global_load_async_to_lds_b32 %0, %1, off"
               :
               : "v"((uint32_t)(uintptr_t)lds_dst),
                 "v"((uint64_t)(uintptr_t)gsrc)
               : "memory");
#endif
}

DEVINL void wait_async0() {
#if __has_builtin(__builtin_amdgcn_s_wait_asynccnt)
  __builtin_amdgcn_s_wait_asynccnt(0);
#else
  asm volatile("s_wait_asynccnt 0x0" ::: "memory");
#endif
  asm volatile("" ::: "memory");  // compiler fence: async engine wrote LDS
}

// LDS tile geometry: 36x36 input tile stored with row stride 48 floats so the
// third WMMA N-chunk (cols 32..47) stays in-bounds; garbage cols 36..47 only
// ever contaminate output columns we never consume (dot products are column-
// isolated, so even NaN garbage cannot leak into consumed outputs).
#define TA_STRIDE 48
#define TA_ELEMS  (36 * TA_STRIDE)   // 1728 floats
#define V_ELEMS   (16 * TA_STRIDE)   // 768 floats

// ---------------------------------------------------------------------------
// One wave per 16x16 tile of the downsampled output.
//   TWO=true : Tin = x - target (level 0)       TWO=false : Tin = d-level
// Produces: outD tile (blurred+downsampled d), and atomically accumulates
// sum |Tin - upsample(outD)| over the tile's 32x32 footprint into accSlot.
// ---------------------------------------------------------------------------
template <bool TWO>
__global__ __launch_bounds__(32)
void pyr_level_kernel(const float* __restrict__ inA,
                      const float* __restrict__ inB,
                      const float* __restrict__ kbuf,
                      float* __restrict__ outD,
                      double* __restrict__ accSlot,
                      int S) {
  extern __shared__ float smem[];
  float* ta   = smem;                    // 36x48 input tile (d values)
  float* vbuf = smem + TA_ELEMS;         // 16x48 vertical-pass result
  float* tb   = smem + TA_ELEMS + V_ELEMS;  // 36x48 target tile (TWO only)

  const int lane = threadIdx.x;   // 0..31 (wave32)
  const int half = lane >> 4;     // 0 | 1
  const int lm   = lane & 15;
  const int tc = blockIdx.x, tr = blockIdx.y, img = blockIdx.z;
  const int S2 = S >> 1;

  const size_t imgoff = (size_t)img * S * S;
  const float* srcA = inA + imgoff;
  const float* srcB = TWO ? (inB + imgoff) : nullptr;
  float* dst = outD + (size_t)img * S2 * S2;

  // 1-D taps from the (separable) 5x5 kernel buffer: K[i][j] = g[i]*g[j],
  // sum(g)=1  =>  g[j] = K[2][j] / sqrt(K[2][2]).
  float g[5];
  {
    float inv = 1.0f / sqrtf(kbuf[12]);
#pragma unroll
    for (int j = 0; j < 5; ++j) g[j] = kbuf[10 + j] * inv;
  }

  const int r0 = 32 * tr - 2, c0 = 32 * tc - 2;
  const bool border = (tr == 0) | (tc == 0) |
                      (tr == (int)gridDim.y - 1) | (tc == (int)gridDim.x - 1);

  if (!border) {
    // Interior tile: whole 36x36 footprint is in-bounds; no zeroing, no
    // per-lane bounds compares — straight-line async issue.
    for (int t = 0; t < 36; ++t) {
      const float* rowp = srcA + (size_t)(r0 + t) * S + c0;
      float* ldsrow = &ta[t * TA_STRIDE];
      async_cp_b32(rowp + lane, ldsrow + lane);
      if (TWO) {
        const float* rowq = srcB + (size_t)(r0 + t) * S + c0;
        async_cp_b32(rowq + lane, &tb[t * TA_STRIDE] + lane);
        if (lane < 4) async_cp_b32(rowq + 32 + lane, &tb[t * TA_STRIDE] + 32 + lane);
      }
      if (lane < 4) async_cp_b32(rowp + 32 + lane, ldsrow + 32 + lane);
    }
  } else {
    // Border tile: zero for the padded conv halo, then predicated fill.
    float4 z = make_float4(0.f, 0.f, 0.f, 0.f);
    float4* p = (float4*)ta;
    for (int i = lane; i < TA_ELEMS / 4; i += 32) p[i] = z;
    if (TWO) {
      float4* q = (float4*)tb;
      for (int i = lane; i < TA_ELEMS / 4; i += 32) q[i] = z;
    }
    __syncthreads();  // zeros committed (dscnt) before async engine writes LDS

    for (int t = 0; t < 36; ++t) {
      int gr = r0 + t;
      if ((unsigned)gr < (unsigned)S) {
        size_t rowbase = (size_t)gr * S;
        int gc = c0 + lane;
        if ((unsigned)gc < (unsigned)S) {
          async_cp_b32(srcA + rowbase + gc, &ta[t * TA_STRIDE + lane]);
          if (TWO) async_cp_b32(srcB + rowbase + gc, &tb[t * TA_STRIDE + lane]);
        }
        if (lane < 4) {
          int gc2 = c0 + 32 + lane;
          if ((unsigned)gc2 < (unsigned)S) {
            async_cp_b32(srcA + rowbase + gc2, &ta[t * TA_STRIDE + 32 + lane]);
            if (TWO) async_cp_b32(srcB + rowbase + gc2, &tb[t * TA_STRIDE + 32 + lane]);
          }
        }
      }
    }
  }
  wait_async0();
  __syncthreads();

  if (TWO) {  // Tin = x - target (halo stays 0 - 0 = 0), b128 LDS ops
    float4* pa = (float4*)ta;
    const float4* pb = (const float4*)tb;
    for (int i = lane; i < TA_ELEMS / 4; i += 32) {
      float4 a = pa[i], b = pb[i];
      a.x -= b.x; a.y -= b.y; a.z -= b.z; a.w -= b.w;
      pa[i] = a;
    }
    __syncthreads();
  }

  // ---- banded coefficient operands (identical for both conv stages) -------
  // A layout (16x4 f32): lane holds M=lm; .x=K(2*half), .y=K(2*half+1).
  // B layout (4x16 f32): lane holds N=lm; .x=K(2*half), .y=K(2*half+1).
  // Slot k - 2*lm for .x is always EVEN -> only {g0,g2,g4}; .y odd -> {g1,g3}.
  // Share the three even-compare results between both components.
  v2f aw[9];
#pragma unroll
  for (int kk = 0; kk < 9; ++kk) {
    int e = 4 * kk + 2 * half - 2 * lm;  // even
    bool e0 = (e == 0), e2 = (e == 2), e4 = (e == 4);
    aw[kk].x = e0 ? g[0] : (e2 ? g[2] : (e4 ? g[4] : 0.0f));
    aw[kk].y = e0 ? g[1] : (e2 ? g[3] : 0.0f);
  }

  // ---- Stage A: V(16x36) = A_band(16x36) x Tin(36x36): 3 N-chunks x 9 K ---
  v8f vacc[3];
#pragma unroll
  for (int nn = 0; nn < 3; ++nn) {
    v8f acc = {};
#pragma unroll
    for (int kk = 0; kk < 9; ++kk) {
      int row = 4 * kk + 2 * half;
      v2f b;
      b.x = ta[row * TA_STRIDE + 16 * nn + lm];
      b.y = ta[(row + 1) * TA_STRIDE + 16 * nn + lm];
      acc = __builtin_amdgcn_wmma_f32_16x16x4_f32(
          false, aw[kk], false, b, (short)0, acc, false, false);
    }
    vacc[nn] = acc;
  }

  // D layout -> LDS (row m = i + 8*half, col = 16*nn + lm)
#pragma unroll
  for (int nn = 0; nn < 3; ++nn)
#pragma unroll
    for (int i = 0; i < 8; ++i)
      vbuf[(i + 8 * half) * TA_STRIDE + 16 * nn + lm] = vacc[nn][i];
  __syncthreads();

  // ---- Stage B: O(16x16) = V(16x36) x B_band(36x16): 9 K-chunks -----------
  v8f oacc = {};
#pragma unroll
  for (int kk = 0; kk < 9; ++kk) {
    int col = 4 * kk + 2 * half;               // even -> 8B aligned
    v2f a = *(const v2f*)(vbuf + lm * TA_STRIDE + col);
    oacc = __builtin_amdgcn_wmma_f32_16x16x4_f32(
        false, a, false, aw[kk], (short)0, oacc, false, false);
  }

  // ---- write next level + Laplacian |d - up(down)| partial sum ------------
  float psum = 0.0f;
#pragma unroll
  for (int i = 0; i < 8; ++i) {
    int m = i + 8 * half;
    float o = oacc[i];
    dst[(size_t)(16 * tr + m) * S2 + 16 * tc + lm] = o;
    int rr = 2 * m + 2, cc = 2 * lm + 2;       // cc even -> aligned v2f loads
    v2f d0 = *(const v2f*)(ta + rr * TA_STRIDE + cc);
    v2f d1 = *(const v2f*)(ta + (rr + 1) * TA_STRIDE + cc);
    psum += fabsf(d0.x - o) + fabsf(d0.y - o) + fabsf(d1.x - o) + fabsf(d1.y - o);
  }
#pragma unroll
  for (int off = 16; off; off >>= 1) psum += __shfl_xor(psum, off, 32);
  if (lane == 0) atomicAdd(accSlot, (double)psum);
}

// ---- residual level: sum |d3| ---------------------------------------------
__global__ __launch_bounds__(256)
void abs_sum_kernel(const float* __restrict__ p, int n, double* __restrict__ acc) {
  int tid = blockIdx.x * blockDim.x + threadIdx.x;
  int stride = gridDim.x * blockDim.x;
  float s = 0.0f;
  for (int i = tid; i < n; i += stride) s += fabsf(p[i]);
#pragma unroll
  for (int off = 16; off; off >>= 1) s += __shfl_xor(s, off, 32);
  if ((threadIdx.x & 31) == 0) atomicAdd(acc, (double)s);
}

__global__ void init_acc_kernel(double* acc) {
  if (threadIdx.x < 4) acc[threadIdx.x] = 0.0;
}

__global__ void finalize_kernel(const double* __restrict__ acc, float* __restrict__ out) {
  if (threadIdx.x == 0) {
    const double BC = 16.0 * 13.0;  // batch * channels = 208
    double loss = acc[0] / (BC * 512.0 * 512.0)
                + acc[1] / (BC * 256.0 * 256.0)
                + acc[2] / (BC * 128.0 * 128.0)
                + acc[3] / (BC * 64.0 * 64.0);
    out[0] = (float)loss;
  }
}

// ---------------------------------------------------------------------------
extern "C" void kernel_launch(void* const* d_in, const int* in_sizes, int n_in,
                              void* d_out, int out_size, void* d_ws, size_t ws_size,
                              hipStream_t stream) {
  (void)in_sizes; (void)n_in; (void)out_size; (void)ws_size;
  const float* x    = (const float*)d_in[0];
  const float* tgt  = (const float*)d_in[1];
  const float* kbuf = (const float*)d_in[2];
  float* out = (float*)d_out;

  const int NIMG = 16 * 13;  // 208 depthwise images
  double* acc = (double*)d_ws;                              // 4 doubles
  float* d1 = (float*)((char*)d_ws + 256);                  // 208*256*256
  float* d2 = d1 + (size_t)NIMG * 256 * 256;                // 208*128*128
  float* d3 = d2 + (size_t)NIMG * 128 * 128;                // 208* 64* 64
  // total ws use: 256 + ~71.6 MB

  const size_t SMEM_TWO = (TA_ELEMS + V_ELEMS + TA_ELEMS) * sizeof(float); // 16896
  const size_t SMEM_ONE = (TA_ELEMS + V_ELEMS) * sizeof(float);            //  9984

  init_acc_kernel<<<1, 32, 0, stream>>>(acc);

  pyr_level_kernel<true ><<<dim3(16, 16, NIMG), dim3(32), SMEM_TWO, stream>>>(
      x, tgt, kbuf, d1, acc + 0, 512);
  pyr_level_kernel<false><<<dim3( 8,  8, NIMG), dim3(32), SMEM_ONE, stream>>>(
      d1, nullptr, kbuf, d2, acc + 1, 256);
  pyr_level_kernel<false><<<dim3( 4,  4, NIMG), dim3(32), SMEM_ONE, stream>>>(
      d2, nullptr, kbuf, d3, acc + 2, 128);

  abs_sum_kernel<<<832, 256, 0, stream>>>(d3, NIMG * 64 * 64, acc + 3);
  finalize_kernel<<<1, 32, 0, stream>>>(acc, out);
}